// GPLightActor_44702019617437
// MI455X (gfx1250) — compile-verified
//
#include <hip/hip_runtime.h>
#include <stdint.h>

#define N_ROWS 262144
#define DIM    128
#define HID    64
#define NG     4
#define NP     4

typedef __attribute__((ext_vector_type(2))) float v2f;
typedef __attribute__((ext_vector_type(8))) float v8f;
typedef __attribute__((ext_vector_type(4))) unsigned int v4u;
typedef __attribute__((ext_vector_type(4))) int v4i;
typedef __attribute__((ext_vector_type(8))) int v8i;

// 6-arg toolchain variant: (uint32x4 g0, int32x8 g1, int32x4 g2, int32x4 g3,
//                           int32x8 extra, i32 cpol) — extra group zeroed.
static __device__ __forceinline__ void tdm_load(v4u g0, v8i g1, v4i g2, v4i g3) {
    v8i gz = {0, 0, 0, 0, 0, 0, 0, 0};
    __builtin_amdgcn_tensor_load_to_lds(g0, g1, g2, g3, gz, 0);
}

// ---------------- workspace layout (uint32 view) ----------------
// w[0..3] counts, w[4..7] cursors, w[16..16+N) compacted row indices
#define WS_IDX_OFF 16
#define WS_NEEDED_BYTES ((size_t)(WS_IDX_OFF + N_ROWS) * 4)

__global__ void k_zero(uint32_t* w) {
    if (threadIdx.x < 16) w[threadIdx.x] = 0u;
}
__global__ void k_count(const long long* __restrict__ gids, uint32_t* w) {
    int i = blockIdx.x * blockDim.x + threadIdx.x;
    if (i < N_ROWS) atomicAdd(&w[(int)gids[i] & 3], 1u);
}
__global__ void k_prefix(uint32_t* w) {
    if (threadIdx.x == 0) {
        w[4] = 0u; w[5] = w[0]; w[6] = w[0] + w[1]; w[7] = w[0] + w[1] + w[2];
    }
}
__global__ void k_scatter(const long long* __restrict__ gids, uint32_t* w) {
    int i = blockIdx.x * blockDim.x + threadIdx.x;
    if (i < N_ROWS) {
        int g = (int)gids[i] & 3;
        uint32_t pos = atomicAdd(&w[4 + g], 1u);
        w[WS_IDX_OFF + pos] = (uint32_t)i;
    }
}

// ---------------- main grouped-GEMM kernel ----------------
// 256 threads = 8 wave32s, one 16-row tile per wave.
// Transposed, bank-padded LDS layouts so every WMMA A/B fragment is one
// contiguous ds_load_b64 (stride 64+4 floats: lane halves k0 / k0+2 start on
// banks 4n+c vs 4n+c+2 -> conflict-free across all 32 lanes).
#define W1T_STRIDE 132               // [g][h][d], d-stride padded
#define W2T_STRIDE 68                // [g][p16][h], h-stride padded
#define HID_STRIDE 68                // [m][h]
#define A_STRIDE   132               // [m][d], matches TDM pad (128+4 dwords)

#define SW1T_FLOATS (NG * HID * W1T_STRIDE)   // 33792
#define SB1_FLOATS  (NG * HID)                // 256
#define SW2T_FLOATS (NG * 16 * W2T_STRIDE)    // 4352
#define SB2_FLOATS  (NG * 16)                 // 64
#define SA_FLOATS   (8 * 16 * A_STRIDE)       // 16896
#define SHID_FLOATS (8 * 16 * HID_STRIDE)     // 8704
#define SMEM_BYTES ((SW1T_FLOATS + SB1_FLOATS + SW2T_FLOATS + SB2_FLOATS + \
                     SA_FLOATS + SHID_FLOATS) * 4 + (8 * 16 * 2) * 4)

__global__ __launch_bounds__(256, 1) void k_mlp(
    const float* __restrict__ hin,
    const long long* __restrict__ gids,
    const unsigned char* __restrict__ fmask,
    const float* __restrict__ W1,
    const float* __restrict__ b1,
    const float* __restrict__ W2,
    const float* __restrict__ b2,
    const uint32_t* __restrict__ idxbuf,   // nullptr => identity ordering
    float* __restrict__ out)
{
    extern __shared__ float smem[];
    float* sW1t = smem;
    float* sB1  = sW1t + SW1T_FLOATS;
    float* sW2t = sB1 + SB1_FLOATS;
    float* sB2  = sW2t + SW2T_FLOATS;
    float* sA   = sB2 + SB2_FLOATS;
    float* sHid = sA + SA_FLOATS;
    int*   sGid = (int*)(sHid + SHID_FLOATS);
    int*   sRow = sGid + 8 * 16;

    const int t    = threadIdx.x;
    const int lane = t & 31;
    const int wave = t >> 5;
    const int tile = blockIdx.x * 8 + wave;

    // per-lane copy of this wave's 16 rows + group ids (for epilogue)
    if (lane < 16) {
        int pos = tile * 16 + lane;
        int row = idxbuf ? (int)idxbuf[WS_IDX_OFF + pos] : pos;
        sRow[wave * 16 + lane] = row;
        sGid[wave * 16 + lane] = (int)gids[row] & 3;
    }

    // ---- TDM gather: DMA this wave's 16 scattered rows (512B each) into the
    // wave-private A staging area, 8 rows per descriptor (32-bit indices),
    // hardware pad 4 dwords per 128 dwords -> LDS row stride 132 floats.
    {
        int base = tile * 16;
        int r[16];
#pragma unroll
        for (int j = 0; j < 16; ++j)           // uniform scalar loads
            r[j] = idxbuf ? (int)idxbuf[WS_IDX_OFF + base + j] : base + j;

        uint64_t gbase = (uint64_t)(uintptr_t)hin;
        uint32_t ldsA  = (uint32_t)(uintptr_t)(sA + wave * 16 * A_STRIDE);

        v4u g0;
        g0[0] = 0x1u | (1u << 30) | (1u << 31);           // count=1, 32b idx, gather
        g0[1] = ldsA;
        g0[2] = (uint32_t)gbase;
        g0[3] = (uint32_t)((gbase >> 32) & 0x01FFFFFFu) | (2u << 30);  // type=2
        v8i g1;
        g1[0] = (int)((2u << 16) | (1u << 20) | (6u << 22) | (3u << 25));
        //            data=4B      pad_en       every 128dw     +4 dwords
        g1[1] = (int)((uint32_t)DIM << 16);               // tensor_dim0 = 128
        g1[2] = 0;                                        // dim0_hi, dim1_lo(262144)=0
        g1[3] = (int)(0x4u | ((uint32_t)DIM << 16));      // dim1_hi=4, tile_dim0=128
        g1[4] = 8;                                        // tile_dim1 = #indices
        g1[5] = DIM;                                      // tensor_dim0_stride = 128
        g1[6] = 0; g1[7] = 0;

        v4i g2a = { r[0], r[1], r[2],  r[3]  };
        v4i g3a = { r[4], r[5], r[6],  r[7]  };
        tdm_load(g0, g1, g2a, g3a);
        g0[1] = ldsA + 8 * A_STRIDE * 4;
        v4i g2b = { r[8],  r[9],  r[10], r[11] };
        v4i g3b = { r[12], r[13], r[14], r[15] };
        tdm_load(g0, g1, g2b, g3b);
    }

    // ---- stage weights (fp32, exact), transposed + padded ----
    for (int i = t; i < NG * DIM * HID; i += 256) {        // W1 [g][d][h] -> [g][h][d]
        int g = i >> 13, r = i & 8191, d = r >> 6, hh = r & 63;
        sW1t[(g * HID + hh) * W1T_STRIDE + d] = W1[i];
    }
    sB1[t] = b1[t];                                        // 256 == 4*64
    for (int i = t; i < SW2T_FLOATS; i += 256) {           // W2 [g][h][p] -> [g][p16][h]
        int g   = i / (16 * W2T_STRIDE);
        int rem = i - g * (16 * W2T_STRIDE);
        int p   = rem / W2T_STRIDE;
        int hh  = rem - p * W2T_STRIDE;
        sW2t[i] = (p < NP && hh < HID) ? W2[(g * HID + hh) * NP + p] : 0.0f;
    }
    if (t < SB2_FLOATS) {
        int g = t >> 4, p = t & 15;
        sB2[t] = (p < NP) ? b2[g * NP + p] : 0.0f;
    }
    __syncthreads();
    __builtin_amdgcn_s_wait_tensorcnt(0);   // wave-private A tile ready

    // groups present in this wave's rows (wave-uniform -> EXEC stays all-ones)
    int gm = 0;
    for (int r = 0; r < 16; ++r) gm |= 1 << sGid[wave * 16 + r];

    const int mrow = lane & 15;
    const int half = lane >> 4;
    const int koff = half << 1;              // K 0..1 vs K 2..3

    const float* aW   = sA + (wave * 16 + mrow) * A_STRIDE + koff;
    float*       hidW = sHid + wave * 16 * HID_STRIDE;
    const v8f vzero = {0.f, 0.f, 0.f, 0.f, 0.f, 0.f, 0.f, 0.f};

    for (int g = 0; g < NG; ++g) {
        if (!((gm >> g) & 1)) continue;

        // ---- layer 1: [16x128] x [128x64], 4 N-tiles x 32 K-steps ----
        const float* bp0 = sW1t + (g * HID +  0 + mrow) * W1T_STRIDE + koff;
        const float* bp1 = sW1t + (g * HID + 16 + mrow) * W1T_STRIDE + koff;
        const float* bp2 = sW1t + (g * HID + 32 + mrow) * W1T_STRIDE + koff;
        const float* bp3 = sW1t + (g * HID + 48 + mrow) * W1T_STRIDE + koff;
        v8f acc[4];
#pragma unroll
        for (int tt = 0; tt < 4; ++tt) acc[tt] = vzero;
#pragma unroll
        for (int kk = 0; kk < 32; ++kk) {
            v2f a = *(const v2f*)(aW + 4 * kk);
            acc[0] = __builtin_amdgcn_wmma_f32_16x16x4_f32(
                false, a, false, *(const v2f*)(bp0 + 4 * kk), (short)0, acc[0], false, false);
            acc[1] = __builtin_amdgcn_wmma_f32_16x16x4_f32(
                false, a, false, *(const v2f*)(bp1 + 4 * kk), (short)0, acc[1], false, false);
            acc[2] = __builtin_amdgcn_wmma_f32_16x16x4_f32(
                false, a, false, *(const v2f*)(bp2 + 4 * kk), (short)0, acc[2], false, false);
            acc[3] = __builtin_amdgcn_wmma_f32_16x16x4_f32(
                false, a, false, *(const v2f*)(bp3 + 4 * kk), (short)0, acc[3], false, false);
        }

        // bias + ReLU, spill hidden[M][h] to LDS to re-fragment for layer 2
#pragma unroll
        for (int tt = 0; tt < 4; ++tt) {
            float bias = sB1[g * HID + tt * 16 + mrow];
#pragma unroll
            for (int r = 0; r < 8; ++r) {
                float x = acc[tt][r] + bias;
                x = x > 0.0f ? x : 0.0f;
                hidW[(r + (half << 3)) * HID_STRIDE + tt * 16 + mrow] = x;
            }
        }
        // same-wave DS ops are architecturally in-order

        // ---- layer 2: [16x64] x [64x16(P padded)] : 16 K-steps ----
        v8f acc2 = vzero;
        const float* w2p = sW2t + (g * 16 + mrow) * W2T_STRIDE + koff;
        const float* a2p = hidW + mrow * HID_STRIDE + koff;
#pragma unroll
        for (int kk = 0; kk < 16; ++kk) {
            acc2 = __builtin_amdgcn_wmma_f32_16x16x4_f32(
                false, *(const v2f*)(a2p + 4 * kk), false,
                *(const v2f*)(w2p + 4 * kk), (short)0, acc2, false, false);
        }
        {
            float bias2 = sB2[g * 16 + mrow];
#pragma unroll
            for (int r = 0; r < 8; ++r)
                hidW[(r + (half << 3)) * 17 + mrow] = acc2[r] + bias2;  // logits
        }

        // ---- epilogue: mask + softmax for rows of group g ----
        if (lane < 16 && sGid[wave * 16 + lane] == g) {
            int row = sRow[wave * 16 + lane];
            const unsigned char* fm = fmask + (size_t)row * NP;
            float x[NP];
#pragma unroll
            for (int p = 0; p < NP; ++p)
                x[p] = fm[p] ? hidW[lane * 17 + p] : -1.0e9f;
            float mx = fmaxf(fmaxf(x[0], x[1]), fmaxf(x[2], x[3]));
            float e0 = expf(x[0] - mx), e1 = expf(x[1] - mx);
            float e2 = expf(x[2] - mx), e3 = expf(x[3] - mx);
            float inv = 1.0f / (e0 + e1 + e2 + e3);
            float4 o = make_float4(e0 * inv, e1 * inv, e2 * inv, e3 * inv);
            *(float4*)(out + (size_t)row * NP) = o;
        }
    }
}

extern "C" void kernel_launch(void* const* d_in, const int* in_sizes, int n_in,
                              void* d_out, int out_size, void* d_ws, size_t ws_size,
                              hipStream_t stream) {
    (void)in_sizes; (void)n_in; (void)out_size;
    const float*         hin   = (const float*)d_in[0];
    const long long*     gids  = (const long long*)d_in[1];
    const unsigned char* fmask = (const unsigned char*)d_in[2];
    const float*         W1    = (const float*)d_in[3];
    const float*         b1    = (const float*)d_in[4];
    const float*         W2    = (const float*)d_in[5];
    const float*         b2    = (const float*)d_in[6];
    float*               outp  = (float*)d_out;
    uint32_t*            w     = (uint32_t*)d_ws;

    (void)hipFuncSetAttribute((const void*)k_mlp,
                              hipFuncAttributeMaxDynamicSharedMemorySize, SMEM_BYTES);

    const uint32_t* idxbuf = nullptr;
    if (ws_size >= WS_NEEDED_BYTES) {
        k_zero<<<1, 32, 0, stream>>>(w);
        k_count<<<N_ROWS / 256, 256, 0, stream>>>(gids, w);
        k_prefix<<<1, 32, 0, stream>>>(w);
        k_scatter<<<N_ROWS / 256, 256, 0, stream>>>(gids, w);
        idxbuf = (const uint32_t*)w;
    }
    k_mlp<<<N_ROWS / (16 * 8), 256, SMEM_BYTES, stream>>>(
        hin, gids, fmask, W1, b1, W2, b2, idxbuf, outp);
}